// CRA_46797963657479
// MI455X (gfx1250) — compile-verified
//
#include <hip/hip_runtime.h>
#include <math.h>

// ---------------------------------------------------------------------------
// Problem: B=256, N=196 tokens, C=192 channels, L=2N=392.
// After algebraic collapse (A symmetric, w5 pushed through w3/w4):
//   phi = relu(BN-folded pointwise conv)            -> 2 GEMMs per batch (WMMA)
//   s[b,c] = sum_m u4[m] * phi[b,c,m]
//   W[b,m] = (u3[c]+s[b,c]) . phi[b,:,m] + K
//   out[b,c,n] = x[b,n,c]*W[b,n] + y[b,n,c]*W[b,n+196]
// Workspace requirement: ~78 MB (cat_phi 77 MB + folded weights) - L2 resident.
// ---------------------------------------------------------------------------

typedef __attribute__((ext_vector_type(2))) float v2f;
typedef __attribute__((ext_vector_type(8))) float v8f;

#define CCH   192
#define NTOK  196
#define LCOL  392

// float offsets inside workspace
#define OFF_W1F 0
#define OFF_W2F 36864
#define OFF_B1F 73728
#define OFF_B2F 73920
#define OFF_U3  74112
#define OFF_U4  74304
#define OFF_KC  74696
#define OFF_PHI 74752            // cat_phi: [256][192][392] floats

// ------------------------- prep: fold BN into conv -------------------------
__global__ void CRA_prep_fold(const float* __restrict__ w1, const float* __restrict__ b1,
                              const float* __restrict__ g1, const float* __restrict__ be1,
                              const float* __restrict__ m1, const float* __restrict__ v1,
                              const float* __restrict__ w2, const float* __restrict__ b2,
                              const float* __restrict__ g2, const float* __restrict__ be2,
                              const float* __restrict__ m2, const float* __restrict__ v2,
                              float* __restrict__ ws) {
  int idx = blockIdx.x * blockDim.x + threadIdx.x;
  if (idx < 36864) {
    int o = idx / CCH;
    float a = g1[o] * rsqrtf(v1[o] + 1e-5f);
    ws[OFF_W1F + idx] = w1[idx] * a;
  } else if (idx < 73728) {
    int e = idx - 36864;
    int o = e / CCH;
    float a = g2[o] * rsqrtf(v2[o] + 1e-5f);
    ws[OFF_W2F + e] = w2[e] * a;
  } else if (idx < 73920) {
    int o = idx - 73728;
    float a = g1[o] * rsqrtf(v1[o] + 1e-5f);
    ws[OFF_B1F + o] = (b1[o] - m1[o]) * a + be1[o];
  } else if (idx < 74112) {
    int o = idx - 73920;
    float a = g2[o] * rsqrtf(v2[o] + 1e-5f);
    ws[OFF_B2F + o] = (b2[o] - m2[o]) * a + be2[o];
  }
}

// --------------------- prep: contract w5 through w3/w4 ---------------------
__global__ void CRA_prep_u(const float* __restrict__ w3, const float* __restrict__ b3,
                           const float* __restrict__ w4, const float* __restrict__ b4,
                           const float* __restrict__ w5, const float* __restrict__ b5,
                           float* __restrict__ ws) {
  int t = threadIdx.x;
  if (t < 192) {                       // u3[i] = sum_c w5[c]*w3[c,i]
    float acc = 0.f;
    for (int c = 0; c < 192; ++c) acc += w5[c] * w3[c * 192 + t];
    ws[OFF_U3 + t] = acc;
  } else if (t < 584) {                // u4[g] = sum_o w5[192+o]*(w4[o,g]+w4[o,g+392])
    int g = t - 192;
    float acc = 0.f;
    for (int o = 0; o < 784; ++o)
      acc += w5[192 + o] * (w4[o * 784 + g] + w4[o * 784 + g + 392]);
    ws[OFF_U4 + g] = acc;
  } else if (t == 584) {               // K = w5a.b3 + w5b.b4 + b5
    float acc = b5[0];
    for (int c = 0; c < 192; ++c) acc += w5[c] * b3[c];
    for (int o = 0; o < 784; ++o) acc += w5[192 + o] * b4[o];
    ws[OFF_KC] = acc;
  }
}

// ------------- main GEMM: cat_phi via V_WMMA_F32_16X16X4_F32 ---------------
// One block = one batch. 8 waves, 312 tiles (2 halves x 12 Mtiles x 13 Ntiles),
// exactly 39 tiles/wave. K = 192 (48 wmma steps per tile).
// Padded token columns (196..207) use a clamped (valid) address; the garbage
// they produce lands only in D columns >= 196, which are never stored, so no
// mask multiply is needed in the inner loop.
__global__ __launch_bounds__(256) void CRA_phi_wmma(const float* __restrict__ x,
                                                    const float* __restrict__ y,
                                                    float* __restrict__ ws) {
  const int b    = blockIdx.x;
  const int tid  = threadIdx.x;
  const int wave = tid >> 5;
  const int lane = tid & 31;
  const int m    = lane & 15;          // A: row M / B: col N / D: col N
  const int kb   = (lane >> 4) << 1;   // K pair {0,1} or {2,3} by lane half
  const int rhi  = (lane >> 4) << 3;   // D: +8 row offset for lanes 16..31

  float* phi = ws + OFF_PHI + (size_t)b * CCH * LCOL;

  for (int t = wave; t < 312; t += 8) {
    const int half = (t >= 156) ? 1 : 0;
    const int tt   = t - half * 156;
    const int mt   = tt / 13;
    const int nt   = tt - mt * 13;

    const float* Wf = ws + (half ? OFF_W2F : OFF_W1F);
    const float* bf = ws + (half ? OFF_B2F : OFF_B1F);
    const float* Xb = (half ? y : x) + (size_t)b * NTOK * CCH;
    const int colbase = half ? NTOK : 0;

    // A fragment source: Wf[mt*16+m][k+kb .. +1]   (M x K = 16 x 4)
    const float* ap = Wf + (mt * 16 + m) * CCH + kb;
    // B fragment source: B[k][n] = Xb[n][k] ; clamp keeps addresses in-bounds
    const int   ng = nt * 16 + m;
    const float* bp = Xb + (ng < NTOK ? ng : NTOK - 1) * CCH + kb;

    v8f acc = {};
#pragma unroll 8
    for (int k = 0; k < CCH; k += 4) {
      v2f a  = *(const v2f*)(ap + k);
      v2f bv = *(const v2f*)(bp + k);
      acc = __builtin_amdgcn_wmma_f32_16x16x4_f32(
                false, a, false, bv, (short)0, acc, false, false);
    }

    // Epilogue: D VGPR r -> row (mt*16 + r + rhi), col (nt*16 + m). Bias+ReLU.
    // Lane needs 8 consecutive bias values: two b128 loads.
    const float4 bia0 = *(const float4*)(bf + mt * 16 + rhi);
    const float4 bia1 = *(const float4*)(bf + mt * 16 + rhi + 4);
    float biasv[8] = { bia0.x, bia0.y, bia0.z, bia0.w,
                       bia1.x, bia1.y, bia1.z, bia1.w };

    const int colg = nt * 16 + m;
    if (colg < NTOK) {
      const int mcol = colbase + colg;
      float* pr = phi + (size_t)(mt * 16 + rhi) * LCOL + mcol;
#pragma unroll
      for (int r = 0; r < 8; ++r) {
        float val = acc[r] + biasv[r];
        val = val > 0.f ? val : 0.f;
        pr[(size_t)r * LCOL] = val;
      }
    }
  }
}

// ------------------- finish: s, W[m], elementwise blend --------------------
__global__ __launch_bounds__(256) void CRA_finish(const float* __restrict__ x,
                                                  const float* __restrict__ y,
                                                  float* __restrict__ out,
                                                  const float* __restrict__ ws) {
  __shared__ float tl[CCH];
  __shared__ float Wv[LCOL];
  const int b    = blockIdx.x;
  const int tid  = threadIdx.x;
  const int wave = tid >> 5;
  const int lane = tid & 31;
  const float* phi = ws + OFF_PHI + (size_t)b * CCH * LCOL;
  const float* u4  = ws + OFF_U4;

  // t[c] = u3[c] + sum_m u4[m]*phi[c][m]   (one wave per c, lane-parallel m)
  for (int c = wave; c < CCH; c += 8) {
    const float* pr = phi + (size_t)c * LCOL;
    float acc = 0.f;
    for (int mm = lane; mm < LCOL; mm += 32) acc += u4[mm] * pr[mm];
    for (int off = 16; off > 0; off >>= 1) acc += __shfl_down(acc, off, 32);
    if (lane == 0) tl[c] = ws[OFF_U3 + c] + acc;
  }
  __syncthreads();

  // W[m] = K + sum_c t[c]*phi[c][m]
  const float Kc = ws[OFF_KC];
  for (int mm = tid; mm < LCOL; mm += 256) {
    float acc = Kc;
    for (int c = 0; c < CCH; ++c) acc += tl[c] * phi[(size_t)c * LCOL + mm];
    Wv[mm] = acc;
  }
  __syncthreads();

  // out[b,c,n] = x[b,n,c]*W[n] + y[b,n,c]*W[196+n]
  const float* xb = x + (size_t)b * NTOK * CCH;
  const float* yb = y + (size_t)b * NTOK * CCH;
  float* ob = out + (size_t)b * CCH * NTOK;
  for (int e = tid; e < CCH * NTOK; e += 256) {
    const int c = e / NTOK;
    const int n = e - c * NTOK;
    ob[e] = xb[n * CCH + c] * Wv[n] + yb[n * CCH + c] * Wv[NTOK + n];
  }
}

extern "C" void kernel_launch(void* const* d_in, const int* in_sizes, int n_in,
                              void* d_out, int out_size, void* d_ws, size_t ws_size,
                              hipStream_t stream) {
  const float* x   = (const float*)d_in[0];
  const float* y   = (const float*)d_in[1];
  const float* w1  = (const float*)d_in[2];
  const float* b1  = (const float*)d_in[3];
  const float* g1  = (const float*)d_in[4];
  const float* be1 = (const float*)d_in[5];
  const float* m1  = (const float*)d_in[6];
  const float* v1  = (const float*)d_in[7];
  const float* w2  = (const float*)d_in[8];
  const float* b2  = (const float*)d_in[9];
  const float* g2  = (const float*)d_in[10];
  const float* be2 = (const float*)d_in[11];
  const float* m2  = (const float*)d_in[12];
  const float* v2  = (const float*)d_in[13];
  const float* w3  = (const float*)d_in[14];
  const float* b3  = (const float*)d_in[15];
  const float* w4  = (const float*)d_in[16];
  const float* b4  = (const float*)d_in[17];
  const float* w5  = (const float*)d_in[18];
  const float* b5  = (const float*)d_in[19];
  float* out = (float*)d_out;
  float* ws  = (float*)d_ws;

  CRA_prep_fold<<<(74112 + 255) / 256, 256, 0, stream>>>(
      w1, b1, g1, be1, m1, v1, w2, b2, g2, be2, m2, v2, ws);
  CRA_prep_u<<<1, 640, 0, stream>>>(w3, b3, w4, b4, w5, b5, ws);
  CRA_phi_wmma<<<256, 256, 0, stream>>>(x, y, ws);
  CRA_finish<<<256, 256, 0, stream>>>(x, y, out, ws);
}